// MarginDiff_24584392802985
// MI455X (gfx1250) — compile-verified
//
#include <hip/hip_runtime.h>

typedef __attribute__((ext_vector_type(2))) float v2f;
typedef __attribute__((ext_vector_type(8))) float v8f;

#define N_PTS  8192
#define DIMS   128
#define KSTEPS (DIMS / 4)      // 32 WMMA k-steps of 4
#define TILE   16
#define NTILES (N_PTS / TILE)  // 512
#define WAVES  8               // 256 threads = 8 wave32
#define TAU_F  1.0f

__global__ void MarginDiff_init_out(float* out) { out[0] = 0.0f; }

__global__ __launch_bounds__(WAVES * 32)
void MarginDiff_fused_kernel(const float* __restrict__ P,
                             const int*   __restrict__ Y,
                             float*       __restrict__ out)
{
    __shared__ float ldsMin[WAVES][TILE];
    __shared__ float ldsMax[WAVES][TILE];

    const int tid  = threadIdx.x;
    const int w    = tid >> 5;      // wave id in block
    const int lane = tid & 31;
    const int half = lane >> 4;     // 0: lanes 0-15, 1: lanes 16-31
    const int l16  = lane & 15;
    const int koff = half * 2;      // K sub-offset per WMMA f32 A/B layout

    const int iBase = blockIdx.x * TILE;
    const int aRow  = iBase + l16;

    // ---- A tile (16 x 128 f32) resident in registers in WMMA layout ----
    // A 16x4: VGPR0 = K=koff, VGPR1 = K=koff+1 for row M = lane%16.
    v2f a[KSTEPS];
    const float* arow = P + (size_t)aRow * DIMS + koff;
#pragma unroll
    for (int kk = 0; kk < KSTEPS; ++kk)
        a[kk] = *(const v2f*)(arow + kk * 4);   // 8B-aligned GLOBAL_LOAD_B64

    // Labels of the 8 rows this lane's C accumulators cover (M = v + half*8)
    int yi[8];
#pragma unroll
    for (int v = 0; v < 8; ++v)
        yi[v] = Y[iBase + v + half * 8];

    const float INF = __builtin_inff();
    float accMin[8], accMax[8];
#pragma unroll
    for (int v = 0; v < 8; ++v) { accMin[v] = INF; accMax[v] = -INF; }

    // ---- sweep all 512 column tiles; each wave takes 2 adjacent tiles per
    // iteration so two independent WMMA accumulator chains can interleave ----
    for (int jt = w * 2; jt < NTILES; jt += WAVES * 2) {
        const int col0 = jt * TILE + l16;           // tile jt   : lane's N column
        const int col1 = col0 + TILE;               // tile jt+1 : lane's N column
        const int yj0  = Y[col0];
        const int yj1  = Y[col1];
        const float* brow0 = P + (size_t)col0 * DIMS + koff;  // B = P^T tiles
        const float* brow1 = P + (size_t)col1 * DIMS + koff;

        v8f c0 = {0.f, 0.f, 0.f, 0.f, 0.f, 0.f, 0.f, 0.f};
        v8f c1 = {0.f, 0.f, 0.f, 0.f, 0.f, 0.f, 0.f, 0.f};
#pragma unroll
        for (int kk = 0; kk < KSTEPS; ++kk) {
            v2f b0 = *(const v2f*)(brow0 + kk * 4);
            v2f b1 = *(const v2f*)(brow1 + kk * 4);
            // Two independent D=A*B+C chains: v_wmma_f32_16x16x4_f32
            c0 = __builtin_amdgcn_wmma_f32_16x16x4_f32(
                     false, a[kk], false, b0, (short)0, c0, false, false);
            c1 = __builtin_amdgcn_wmma_f32_16x16x4_f32(
                     false, a[kk], false, b1, (short)0, c1, false, false);
        }

        // Masked accumulate: element (i = iBase+v+half*8, j = col) = c[v]
#pragma unroll
        for (int v = 0; v < 8; ++v) {
            const int iRow = iBase + v + half * 8;

            const float s0    = c0[v];
            const bool  same0 = (yi[v] == yj0) && (iRow != col0);
            const bool  diff0 = (yi[v] != yj0);
            accMin[v] = fminf(accMin[v], same0 ? s0 :  INF);
            accMax[v] = fmaxf(accMax[v], diff0 ? s0 : -INF);

            const float s1    = c1[v];
            const bool  same1 = (yi[v] == yj1) && (iRow != col1);
            const bool  diff1 = (yi[v] != yj1);
            accMin[v] = fminf(accMin[v], same1 ? s1 :  INF);
            accMax[v] = fmaxf(accMax[v], diff1 ? s1 : -INF);
        }
    }

    // ---- reduce over the 16 N-lanes within each half-wave ----
#pragma unroll
    for (int m = 1; m < 16; m <<= 1) {
#pragma unroll
        for (int v = 0; v < 8; ++v) {
            accMin[v] = fminf(accMin[v], __shfl_xor(accMin[v], m, 32));
            accMax[v] = fmaxf(accMax[v], __shfl_xor(accMax[v], m, 32));
        }
    }
    // lane 0 holds rows 0..7, lane 16 holds rows 8..15
    if (l16 == 0) {
#pragma unroll
        for (int v = 0; v < 8; ++v) {
            ldsMin[w][half * 8 + v] = accMin[v];
            ldsMax[w][half * 8 + v] = accMax[v];
        }
    }
    __syncthreads();

    // ---- combine 8 waves, finalize 16 rows, accumulate mean ----
    if (w == 0 && lane < 16) {
        float mn = INF, mx = -INF;
#pragma unroll
        for (int q = 0; q < WAVES; ++q) {
            mn = fminf(mn, ldsMin[q][lane]);
            mx = fmaxf(mx, ldsMax[q][lane]);
        }
        const float minSame = (mn ==  INF) ? 0.0f : mn;   // row had no same-class peer
        const float maxDiff = (mx == -INF) ? 0.0f : mx;   // row had no diff-class peer
        float loss = fmaxf(0.0f, minSame + maxDiff + TAU_F);
#pragma unroll
        for (int m = 1; m < 16; m <<= 1)
            loss += __shfl_xor(loss, m, 32);              // stays within lanes 0-15
        if (lane == 0)
            atomicAdd(out, loss * (1.0f / (float)N_PTS)); // mean reduction
    }
}

extern "C" void kernel_launch(void* const* d_in, const int* in_sizes, int n_in,
                              void* d_out, int out_size, void* d_ws, size_t ws_size,
                              hipStream_t stream) {
    const float* P = (const float*)d_in[0];   // projection [8192,128] f32
    const int*   Y = (const int*)  d_in[1];   // labels [8192] int
    float* out = (float*)d_out;               // scalar loss

    MarginDiff_init_out<<<1, 1, 0, stream>>>(out);
    MarginDiff_fused_kernel<<<NTILES, WAVES * 32, 0, stream>>>(P, Y, out);
}